// NormalRender_2164663517576
// MI455X (gfx1250) — compile-verified
//
#include <hip/hip_runtime.h>
#include <stdint.h>

// ---------------- problem constants ----------------
#define B_    16
#define V_    6890
#define NF_   13776
#define HW_   (512 * 512)          // H*W, K = 1
#define NPIX_ (B_ * HW_)
#define SIGMA_INV 10000.0f         // 1 / 1e-4

// ---------------- TDM (Tensor Data Mover) helpers ----------------
typedef __attribute__((ext_vector_type(4))) unsigned int u32x4_;
typedef __attribute__((ext_vector_type(8))) int          i32x8_;
typedef __attribute__((ext_vector_type(4))) int          i32x4_;

// 1-D DWORD-tile global->LDS DMA. D# layout per CDNA5 ISA §8:
//   group0: count=1 | lds_addr | global_addr[56:0] | type=2
//   group1: data_size=2 (4B), tensor_dim0=nelem, tensor_dim1=1,
//           tile_dim0=nelem, tile_dim1=1, stride0=nelem
// 6-arg toolchain form: (g0, g1, g2, g3, g_extra, cpol)
__device__ __forceinline__ void tdm_load_1d_dw(uint32_t lds_byte_off,
                                               const void* gptr,
                                               uint32_t nelem) {
  uint64_t ga = (uint64_t)(uintptr_t)gptr;
  u32x4_ g0;
  g0.x = 1u;                                    // count=1 (valid), is_restore=0
  g0.y = lds_byte_off;                          // lds_addr (bytes)
  g0.z = (uint32_t)ga;                          // global_addr[31:0]
  g0.w = (uint32_t)((ga >> 32) & 0x01FFFFFFu)   // global_addr[56:32]
         | 0x80000000u;                         // type=2 ("image"), bits 127:126
  i32x8_ g1;
  g1[0] = (int)(2u << 16);                      // workgroup_mask=0, data_size=2 (4B)
  g1[1] = (int)((nelem & 0xFFFFu) << 16);       // tensor_dim0[15:0] in [31:16]
  g1[2] = (int)((nelem >> 16) & 0xFFFFu)        // tensor_dim0[31:16]
        | (int)(1u << 16);                      // tensor_dim1[15:0] = 1
  g1[3] = (int)((nelem & 0xFFFFu) << 16);       // tile_dim0 = nelem (<= 65535)
  g1[4] = 1;                                    // tile_dim1 = 1, tile_dim2 = 0
  g1[5] = (int)nelem;                           // tensor_dim0_stride[31:0]
  g1[6] = 0;                                    // stride0[47:32], stride1[15:0]
  g1[7] = 0;                                    // stride1[47:16]
  i32x4_ gz4 = {0, 0, 0, 0};                    // groups 2/3 unused (<=2D tensor)
  i32x8_ gz8 = {0, 0, 0, 0, 0, 0, 0, 0};
  __builtin_amdgcn_tensor_load_to_lds(g0, g1, gz4, gz4, gz8, 0);
}

// ---------------- kernel 1: vertex normals + face attr sums ----------------
// One workgroup per batch. Whole per-batch vertex-normal accumulator lives in
// LDS (6890*3*4 = 82,680 B — exploits CDNA5's 320 KB/WGP LDS) with ds_add_f32
// atomics; then normalize in LDS; then emit face_attr_sum straight from LDS.
// FSTRIDE = 4: pad each face record to 16 B so the pixel kernel gathers with a
// single aligned global_load_b128. FSTRIDE = 3: tight fallback.
template <int FSTRIDE>
__global__ __launch_bounds__(256) void nr_vn_fas_kernel(
    const float* __restrict__ verts,   // (B, V, 3)
    const int*   __restrict__ faces,   // (NF, 3)
    float*       __restrict__ fas) {   // (B, NF, FSTRIDE) workspace
  __shared__ float vn[V_ * 3];
  const int b = blockIdx.x;
  const float* vb = verts + (size_t)b * (V_ * 3);

  for (int i = threadIdx.x; i < V_ * 3; i += 256) vn[i] = 0.0f;
  __syncthreads();

  for (int f = threadIdx.x; f < NF_; f += 256) {
    const int i0 = faces[3 * f + 0];
    const int i1 = faces[3 * f + 1];
    const int i2 = faces[3 * f + 2];
    const float ax = vb[3 * i0 + 0], ay = vb[3 * i0 + 1], az = vb[3 * i0 + 2];
    const float bx = vb[3 * i1 + 0], by = vb[3 * i1 + 1], bz = vb[3 * i1 + 2];
    const float cx = vb[3 * i2 + 0], cy = vb[3 * i2 + 1], cz = vb[3 * i2 + 2];
    const float e1x = bx - ax, e1y = by - ay, e1z = bz - az;
    const float e2x = cx - ax, e2y = cy - ay, e2z = cz - az;
    const float nx = e1y * e2z - e1z * e2y;
    const float ny = e1z * e2x - e1x * e2z;
    const float nz = e1x * e2y - e1y * e2x;
    atomicAdd(&vn[3 * i0 + 0], nx); atomicAdd(&vn[3 * i0 + 1], ny); atomicAdd(&vn[3 * i0 + 2], nz);
    atomicAdd(&vn[3 * i1 + 0], nx); atomicAdd(&vn[3 * i1 + 1], ny); atomicAdd(&vn[3 * i1 + 2], nz);
    atomicAdd(&vn[3 * i2 + 0], nx); atomicAdd(&vn[3 * i2 + 1], ny); atomicAdd(&vn[3 * i2 + 2], nz);
  }
  __syncthreads();

  for (int v = threadIdx.x; v < V_; v += 256) {
    const float x = vn[3 * v + 0], y = vn[3 * v + 1], z = vn[3 * v + 2];
    const float inv = 1.0f / fmaxf(sqrtf(x * x + y * y + z * z), 1e-6f);
    vn[3 * v + 0] = x * inv; vn[3 * v + 1] = y * inv; vn[3 * v + 2] = z * inv;
  }
  __syncthreads();

  float* fb = fas + (size_t)b * (NF_ * FSTRIDE);
  for (int f = threadIdx.x; f < NF_; f += 256) {
    const int i0 = faces[3 * f + 0];
    const int i1 = faces[3 * f + 1];
    const int i2 = faces[3 * f + 2];
    const float sx = vn[3 * i0 + 0] + vn[3 * i1 + 0] + vn[3 * i2 + 0];
    const float sy = vn[3 * i0 + 1] + vn[3 * i1 + 1] + vn[3 * i2 + 1];
    const float sz = vn[3 * i0 + 2] + vn[3 * i1 + 2] + vn[3 * i2 + 2];
    if (FSTRIDE == 4) {
      float4 r; r.x = sx; r.y = sy; r.z = sz; r.w = 0.0f;
      ((float4*)fb)[f] = r;                     // single b128 store
    } else {
      fb[3 * f + 0] = sx; fb[3 * f + 1] = sy; fb[3 * f + 2] = sz;
    }
  }
}

// ---------------- kernel 2: per-pixel shading ----------------
// K=1 collapses the compositing chain; zbuf and the softmax weights are dead
// code w.r.t. the output, so they are never read. Streamed inputs (dists,
// pix_to_face) come in per-tile via TDM async copies into LDS; the <=3.5 MB
// face_attr_sum table is gathered through L2 (192 MB).
#define TILE_ 2048                 // 2048 | HW_, so a tile never crosses batches
template <int FSTRIDE>
__global__ __launch_bounds__(256) void nr_pixel_kernel(
    const float* __restrict__ dists,   // (B, H, W, 1)
    const int*   __restrict__ p2f,     // (B, H, W, 1)
    const float* __restrict__ fas,     // (B, NF, FSTRIDE)
    float4*      __restrict__ out) {   // (B, H, W, 4)
  __shared__ float s_d[TILE_];
  __shared__ int   s_f[TILE_];
  const int base = blockIdx.x * TILE_;

  if (threadIdx.x < 32) {               // wave 0 issues the DMAs once
    const uint32_t d_off = (uint32_t)(uintptr_t)(void*)s_d;
    const uint32_t f_off = (uint32_t)(uintptr_t)(void*)s_f;
    tdm_load_1d_dw(d_off, dists + base, TILE_);
    tdm_load_1d_dw(f_off, p2f + base, TILE_);
    __builtin_amdgcn_s_wait_tensorcnt(0);
  }
  __syncthreads();

  const int b = base >> 18;             // HW_ = 2^18
  const float* fb = fas + (size_t)b * (NF_ * FSTRIDE);
  __builtin_prefetch(fb, 0, 3);         // global_prefetch_b8: warm gather table

  for (int i = threadIdx.x; i < TILE_; i += 256) {
    const int   f = s_f[i];
    const float d = s_d[i];
    const bool valid = (f >= 0);
    const int idx = valid ? (f < NF_ ? f : NF_ - 1) : 0;
    float x = 0.0f, y = 0.0f, z = 0.0f;
    if (valid) {
      if (FSTRIDE == 4) {
        const float4 v = ((const float4*)fb)[idx];   // single b128 gather
        x = v.x; y = v.y; z = v.z;
      } else {
        x = fb[3 * idx + 0];
        y = fb[3 * idx + 1];
        z = fb[3 * idx + 2];
      }
    }
    const float n   = sqrtf(x * x + y * y + z * z);
    const float inv = 1.0f / fmaxf(n, 1e-12f);
    const float miss = (n > 0.0f) ? 0.0f : 1.0f;          // 1 - hit
    // prob = sigmoid(-d/sigma) * mask ; rendered_alpha = 1 - prob (K = 1)
    const float prob = valid ? (1.0f / (1.0f + __expf(d * SIGMA_INV))) : 0.0f;
    float4 o;
    o.x = x * inv + miss;
    o.y = y * inv + miss;
    o.z = z * inv + miss;
    o.w = 1.0f - prob;
    out[base + i] = o;
  }
}

// ---------------- launch ----------------
extern "C" void kernel_launch(void* const* d_in, const int* in_sizes, int n_in,
                              void* d_out, int out_size, void* d_ws, size_t ws_size,
                              hipStream_t stream) {
  const float* verts = (const float*)d_in[0];  // mesh_verts (B,V,3)
  // d_in[1] = zbuf: dead code w.r.t. output, intentionally unread
  const float* dists = (const float*)d_in[2];  // (B,H,W,1)
  const int*   faces = (const int*)d_in[3];    // (NF,3)
  const int*   p2f   = (const int*)d_in[4];    // (B,H,W,1)
  float*  fas = (float*)d_ws;                  // (B,NF,{4|3}) scratch
  float4* out = (float4*)d_out;                // (B,H,W,4)

  // Prefer 16B-padded face records (single b128 gather) when scratch allows.
  const bool pad4 = ws_size >= (size_t)B_ * NF_ * 4 * sizeof(float);
  if (pad4) {
    nr_vn_fas_kernel<4><<<dim3(B_), dim3(256), 0, stream>>>(verts, faces, fas);
    nr_pixel_kernel<4><<<dim3(NPIX_ / TILE_), dim3(256), 0, stream>>>(dists, p2f, fas, out);
  } else {
    nr_vn_fas_kernel<3><<<dim3(B_), dim3(256), 0, stream>>>(verts, faces, fas);
    nr_pixel_kernel<3><<<dim3(NPIX_ / TILE_), dim3(256), 0, stream>>>(dists, p2f, fas, out);
  }
}